// UVUTensorProduct_76312978916003
// MI455X (gfx1250) — compile-verified
//
#include <hip/hip_runtime.h>
#include <math.h>

typedef __attribute__((ext_vector_type(2))) float v2f;
typedef __attribute__((ext_vector_type(8))) float v8f;

// ---------------------------------------------------------------------------
// f32 WMMA: D(16x16) = A(16x4) * B(4x16) + C
// A layout: lane L: m=L&15, khalf=L>>4; a.x -> K = 2*khalf, a.y -> K = 2*khalf+1
// B layout: lane L: n=L&15, khalf=L>>4; b.x -> K = 2*khalf, b.y -> K = 2*khalf+1
// C/D layout: reg r, lane L: row m = r + 8*(L>>4), col n = L&15
// ---------------------------------------------------------------------------
static __device__ __forceinline__ v8f wmma4(v2f a, v2f b, v8f c) {
  return __builtin_amdgcn_wmma_f32_16x16x4_f32(
      /*neg_a=*/false, a, /*neg_b=*/false, b,
      /*c_mod=*/(short)0, c, /*reuse_a=*/false, /*reuse_b=*/false);
}

// ===========================================================================
// d_ws layout (dwords):
//   0     : int tab[2*35]  (offset,count per slot; pad to 128)
//   128   : int2 ent[512]  (.x = (i_chan<<4)|j_chan, .y = float bits of coeff)
//   1152  : packed lin0_w  (192 rows)  12288
//   13440 : packed sc_w    (192 rows)  12288
//   25728 : packed lin_w0  (192 rows)  12288
//   38016 : packed lin_w1  (256 rows)  16384
//   54400 : packed lin_w2  (256 rows)  16384   (end = 70784 dw = 283136 B)
// Packed layout: dst[((r>>2)*64 + c)*4 + (r&3)] = src[r*64+c]  -> a lane's
// B-fragment (rows 4t+2kh, 4t+2kh+1 of column c) is one aligned b64 load.
// ===========================================================================
#define WS_ENT   128
#define WS_PL0   1152
#define WS_PSC   13440
#define WS_PW0   25728
#define WS_PW1   38016
#define WS_PW2   54400

__global__ void pack_weights_kernel(const float* __restrict__ src,
                                    float* __restrict__ dst, int rows) {
  int idx = blockIdx.x * 256 + threadIdx.x;
  if (idx >= rows * 64) return;
  int r = idx >> 6, c = idx & 63;
  dst[((r >> 2) * 64 + c) * 4 + (r & 3)] = src[idx];
}

// --------------------- Wigner-3j setup (double precision) ------------------
struct cpx { double re, im; };
static __device__ __forceinline__ cpx cmul(cpx a, cpx b) {
  cpx r; r.re = a.re * b.re - a.im * b.im; r.im = a.re * b.im + a.im * b.re; return r;
}
static __device__ __forceinline__ cpx cconj(cpx a) { cpx r; r.re = a.re; r.im = -a.im; return r; }

static __device__ double dfact(int n) {
  double r = 1.0;
  for (int i = 2; i <= n; ++i) r *= (double)i;
  return r;
}

static __device__ double su2_cg(int j1, int m1, int j2, int m2, int j3, int m3) {
  if (m3 != m1 + m2) return 0.0;
  int vmin = -j1 + j2 + m3; if (-j1 + m1 > vmin) vmin = -j1 + m1; if (0 > vmin) vmin = 0;
  int vmax = j2 + j3 + m1; if (j3 - j1 + j2 < vmax) vmax = j3 - j1 + j2; if (j3 + m3 < vmax) vmax = j3 + m3;
  double C = sqrt((2.0 * j3 + 1.0) * dfact(j3 + j1 - j2) * dfact(j3 - j1 + j2) * dfact(j1 + j2 - j3) *
                  dfact(j3 + m3) * dfact(j3 - m3) /
                  (dfact(j1 + j2 + j3 + 1) * dfact(j1 - m1) * dfact(j1 + m1) * dfact(j2 - m2) * dfact(j2 + m2)));
  double S = 0.0;
  for (int v = vmin; v <= vmax; ++v) {
    double sgn = ((v + j2 + m2) & 1) ? -1.0 : 1.0;
    S += sgn * dfact(j2 + j3 + m1 - v) * dfact(j1 - m1 + v) /
         (dfact(v) * dfact(j3 - j1 + j2 - v) * dfact(j3 + m3 - v) * dfact(v + j1 - j2 - m3));
  }
  return C * S;
}

static __device__ void qmat(int l, cpx q[5][5]) {
  for (int i = 0; i < 5; ++i)
    for (int j = 0; j < 5; ++j) { q[i][j].re = 0.0; q[i][j].im = 0.0; }
  const double s = 0.70710678118654752440;
  for (int mm = -l; mm < 0; ++mm) {
    q[l + mm][l - mm].re = s;
    q[l + mm][l + mm].im = -s;
  }
  q[l][l].re = 1.0;
  for (int mm = 1; mm <= l; ++mm) {
    double sg = (mm & 1) ? -1.0 : 1.0;
    q[l + mm][l + mm].re = sg * s;
    q[l + mm][l - mm].im = sg * s;
  }
  cpx f;
  switch (l & 3) {  // (-1j)^l
    case 0: f.re = 1.0;  f.im = 0.0;  break;
    case 1: f.re = 0.0;  f.im = -1.0; break;
    case 2: f.re = -1.0; f.im = 0.0;  break;
    default: f.re = 0.0; f.im = 1.0;  break;
  }
  for (int i = 0; i < 5; ++i)
    for (int j = 0; j < 5; ++j) q[i][j] = cmul(f, q[i][j]);
}

static __device__ void compute_w3j(int l1, int l2, int l3, double W[5][5][5]) {
  double C[5][5][5];
  for (int i = 0; i < 5; ++i)
    for (int j = 0; j < 5; ++j)
      for (int k = 0; k < 5; ++k) C[i][j][k] = 0.0;
  for (int m1 = -l1; m1 <= l1; ++m1)
    for (int m2 = -l2; m2 <= l2; ++m2) {
      int m3 = m1 + m2;
      if (m3 >= -l3 && m3 <= l3)
        C[l1 + m1][l2 + m2][l3 + m3] = su2_cg(l1, m1, l2, m2, l3, m3);
    }
  double inv = 1.0 / sqrt(2.0 * l3 + 1.0);
  cpx q1[5][5], q2[5][5], q3[5][5];
  qmat(l1, q1); qmat(l2, q2); qmat(l3, q3);
  int n1 = 2 * l1 + 1, n2 = 2 * l2 + 1, n3 = 2 * l3 + 1;
  for (int j = 0; j < n1; ++j)
    for (int lo = 0; lo < n2; ++lo)
      for (int mm = 0; mm < n3; ++mm) {
        double acc = 0.0;
        for (int i = 0; i < n1; ++i)
          for (int kk = 0; kk < n2; ++kk)
            for (int nn = 0; nn < n3; ++nn) {
              double c = C[i][kk][nn] * inv;
              if (c == 0.0) continue;
              cpx t = cmul(q1[i][j], q2[kk][lo]);
              t = cmul(t, cconj(q3[nn][mm]));
              acc += t.re * c;
            }
        W[j][lo][mm] = acc;
      }
}

__global__ void setup_w3j_kernel(int* __restrict__ tab, int2* __restrict__ ent) {
  if (threadIdx.x != 0 || blockIdx.x != 0) return;
  const int pathsL[11][3] = {{0,0,0},{0,1,1},{0,2,2},{1,0,1},{1,1,0},{1,1,2},
                             {1,2,1},{2,0,2},{2,1,1},{2,2,0},{2,2,2}};
  const int groups[3][4] = {{0,4,9,0},{1,3,6,8},{2,5,7,10}};
  const int nb[3] = {3,4,4};
  const int chanbase[3] = {0,1,4};
  int ecount = 0, slot = 0;
  for (int l = 0; l < 3; ++l) {
    double scale = sqrt(2.0 * l + 1.0) / sqrt((double)(nb[l] * 64));
    for (int pl = 0; pl < nb[l]; ++pl) {
      int p = groups[l][pl];
      int li = pathsL[p][0], lj = pathsL[p][1];
      double W[5][5][5];
      compute_w3j(li, lj, l, W);
      for (int k = 0; k < 2 * l + 1; ++k) {
        int c0 = ecount;
        for (int i = 0; i < 2 * li + 1; ++i)
          for (int j = 0; j < 2 * lj + 1; ++j) {
            double v = W[i][j][k] * scale;
            if (fabs(v) > 1e-9) {
              ent[ecount] = make_int2(((chanbase[li] + i) << 4) | (chanbase[lj] + j),
                                      __float_as_int((float)v));
              ++ecount;
            }
          }
        tab[2 * slot] = c0;
        tab[2 * slot + 1] = ecount - c0;
        ++slot;
      }
    }
  }
}

// ===========================================================================
// Fused main kernel: 128 threads = 4 waves, wave handles a 16-node tile.
// LDS: x1 per wave = 16 nodes x (64 u x 9 chans), node stride 577 dwords
// (577 = 64*9+1, odd pad so node-strided mid-build reads are conflict free).
// ===========================================================================
template <int L, int DB, int DC>
static __device__ __forceinline__ void gemm1_chunk(
    const float* __restrict__ arow, const float* __restrict__ pw,
    float* __restrict__ X1, const float* __restrict__ d3row, int lane) {
  constexpr int D = 2 * L + 1;
  constexpr int CH = (L == 0) ? 0 : ((L == 1) ? 1 : 4);
  const int m = lane & 15;
  const int kh = lane >> 4;
  v8f vzero = {};
  v8f acc[DC][4];
#pragma unroll
  for (int d = 0; d < DC; ++d)
#pragma unroll
    for (int wt = 0; wt < 4; ++wt) acc[d][wt] = vzero;
#pragma unroll 2
  for (int t = 0; t < 16; ++t) {
    const int k0 = 4 * t + 2 * kh;
    v2f a[DC];
#pragma unroll
    for (int d = 0; d < DC; ++d) {
      a[d].x = arow[k0 * D + DB + d];
      a[d].y = arow[(k0 + 1) * D + DB + d];
    }
    v2f b[4];
#pragma unroll
    for (int wt = 0; wt < 4; ++wt)
      b[wt] = *(const v2f*)(pw + ((t * 64 + wt * 16 + m) * 4 + 2 * kh));
#pragma unroll
    for (int d = 0; d < DC; ++d)
#pragma unroll
      for (int wt = 0; wt < 4; ++wt) acc[d][wt] = wmma4(a[d], b[wt], acc[d][wt]);
  }
#pragma unroll
  for (int d = 0; d < DC; ++d)
#pragma unroll
    for (int wt = 0; wt < 4; ++wt)
#pragma unroll
      for (int r = 0; r < 8; ++r)
        X1[(r + 8 * kh) * 577 + (wt * 16 + m) * 9 + CH + DB + d] =
            0.125f * d3row[r] * acc[d][wt][r];
}

template <int L, int KB, int KC>
static __device__ __forceinline__ void gemm2_chunk(
    const float* __restrict__ scrow, const float* __restrict__ wrow,
    const float* __restrict__ pw_lw, const float* __restrict__ pw_sc,
    const int* __restrict__ tab, const int2* __restrict__ ent,
    const float* __restrict__ xrow, const float* __restrict__ ereg,
    const float* __restrict__ d3row, float* __restrict__ out,
    int nodeBase, int lane) {
  constexpr int D = 2 * L + 1;
  constexpr int NB = (L == 0) ? 3 : 4;
  constexpr int SLOTBASE = (L == 0) ? 0 : ((L == 1) ? 3 : 15);
  constexpr int OBASE = (L == 0) ? 0 : ((L == 1) ? 64 : 256);
  const int m = lane & 15;
  const int kh = lane >> 4;
  v8f vzero = {};
  v8f acc[KC][4];
#pragma unroll
  for (int kk = 0; kk < KC; ++kk)
#pragma unroll
    for (int wt = 0; wt < 4; ++wt) acc[kk][wt] = vzero;

  // ---- main GEMM over concatenated path blocks (K = NB*64) ----
#pragma unroll 2
  for (int t = 0; t < NB * 16; ++t) {
    const int pl = t >> 4;
    const int pg = (L == 0) ? ((pl == 0) ? 0 : (pl == 1) ? 4 : 9)
                 : (L == 1) ? ((pl == 0) ? 1 : (pl == 1) ? 3 : (pl == 2) ? 6 : 8)
                            : ((pl == 0) ? 2 : (pl == 1) ? 5 : (pl == 2) ? 7 : 10);
    const int ub = ((4 * t) & 63) + 2 * kh;
    const float w0 = wrow[pg * 64 + ub];
    const float w1 = wrow[pg * 64 + ub + 1];
    v2f b[4];
#pragma unroll
    for (int wt = 0; wt < 4; ++wt)
      b[wt] = *(const v2f*)(pw_lw + ((t * 64 + wt * 16 + m) * 4 + 2 * kh));
#pragma unroll
    for (int kk = 0; kk < KC; ++kk) {
      const int k = KB + kk;
      const int slot = SLOTBASE + pl * D + k;
      const int off = tab[2 * slot];
      const int cnt = tab[2 * slot + 1];
      float s0 = 0.0f, s1 = 0.0f;
      for (int e2 = 0; e2 < cnt; ++e2) {
        const int2 ev = ent[off + e2];
        const float evv = __int_as_float(ev.y) * ereg[ev.x & 15];
        const int ich = ev.x >> 4;
        s0 += evv * xrow[ub * 9 + ich];
        s1 += evv * xrow[(ub + 1) * 9 + ich];
      }
      v2f a;
      a.x = w0 * s0;
      a.y = w1 * s1;
#pragma unroll
      for (int wt = 0; wt < 4; ++wt) acc[kk][wt] = wmma4(a, b[wt], acc[kk][wt]);
    }
  }
  // ---- skip connection: (1/8) * data1 . sc_w (data3 applied in epilogue) ----
#pragma unroll 2
  for (int t = 0; t < 16; ++t) {
    const int u0 = 4 * t + 2 * kh;
    v2f b[4];
#pragma unroll
    for (int wt = 0; wt < 4; ++wt)
      b[wt] = *(const v2f*)(pw_sc + ((t * 64 + wt * 16 + m) * 4 + 2 * kh));
#pragma unroll
    for (int kk = 0; kk < KC; ++kk) {
      const int k = KB + kk;
      v2f a;
      a.x = 0.125f * scrow[u0 * D + k];
      a.y = 0.125f * scrow[(u0 + 1) * D + k];
#pragma unroll
      for (int wt = 0; wt < 4; ++wt) acc[kk][wt] = wmma4(a, b[wt], acc[kk][wt]);
    }
  }
  // ---- epilogue: scale rows by data3 and store ----
#pragma unroll
  for (int kk = 0; kk < KC; ++kk)
#pragma unroll
    for (int wt = 0; wt < 4; ++wt)
#pragma unroll
      for (int r = 0; r < 8; ++r) {
        const int mm = r + 8 * kh;
        const int w = wt * 16 + m;
        out[(size_t)(nodeBase + mm) * 576 + OBASE + w * D + KB + kk] =
            d3row[r] * acc[kk][wt][r];
      }
}

__global__ __launch_bounds__(128) void fused_tp_kernel(
    const float* __restrict__ data1, const float* __restrict__ data2,
    const float* __restrict__ dwt, const float* __restrict__ d3,
    const float* __restrict__ ws, float* __restrict__ out) {
  __shared__ float x1buf[4 * 16 * 577];
  const int tid = threadIdx.x;
  const int wave = tid >> 5;
  const int lane = tid & 31;
  const int m = lane & 15;
  const int kh = lane >> 4;
  const int nodeBase = blockIdx.x * 64 + wave * 16;
  float* X1 = x1buf + wave * (16 * 577);

  const int* tab = (const int*)ws;
  const int2* ent = (const int2*)(ws + WS_ENT);
  const float* pl0 = ws + WS_PL0;
  const float* psc = ws + WS_PSC;
  const float* pw0 = ws + WS_PW0;
  const float* pw1 = ws + WS_PW1;
  const float* pw2 = ws + WS_PW2;

  // prefetch this wave's node rows (global_prefetch_b8)
  __builtin_prefetch(data1 + (size_t)(nodeBase + m) * 576, 0, 3);
  __builtin_prefetch(dwt + (size_t)(nodeBase + m) * 704, 0, 3);

  float ereg[9];
#pragma unroll
  for (int j = 0; j < 9; ++j) ereg[j] = data2[(size_t)(nodeBase + m) * 9 + j];
  float d3row[8];
#pragma unroll
  for (int r = 0; r < 8; ++r) d3row[r] = d3[nodeBase + r + 8 * kh];

  const float* arow0 = data1 + (size_t)(nodeBase + m) * 576;
  const float* wrow = dwt + (size_t)(nodeBase + m) * 704;
  const float* xrow = X1 + m * 577;

  // GEMM1: x1 = (data3/8) * (x . lin0_w) -> LDS (per-wave, no barrier needed)
  gemm1_chunk<0, 0, 1>(arow0, pl0, X1, d3row, lane);
  gemm1_chunk<1, 0, 3>(arow0 + 64, pl0 + 4096, X1, d3row, lane);
  gemm1_chunk<2, 0, 3>(arow0 + 256, pl0 + 8192, X1, d3row, lane);
  gemm1_chunk<2, 3, 2>(arow0 + 256, pl0 + 8192, X1, d3row, lane);

  // GEMM2 (+ skip connection) per output irrep, k-chunked
  gemm2_chunk<0, 0, 1>(arow0, wrow, pw0, psc, tab, ent, xrow, ereg, d3row, out, nodeBase, lane);
  gemm2_chunk<1, 0, 3>(arow0 + 64, wrow, pw1, psc + 4096, tab, ent, xrow, ereg, d3row, out, nodeBase, lane);
  gemm2_chunk<2, 0, 3>(arow0 + 256, wrow, pw2, psc + 8192, tab, ent, xrow, ereg, d3row, out, nodeBase, lane);
  gemm2_chunk<2, 3, 2>(arow0 + 256, wrow, pw2, psc + 8192, tab, ent, xrow, ereg, d3row, out, nodeBase, lane);
}

// ===========================================================================
extern "C" void kernel_launch(void* const* d_in, const int* in_sizes, int n_in,
                              void* d_out, int out_size, void* d_ws, size_t ws_size,
                              hipStream_t stream) {
  (void)n_in; (void)out_size; (void)ws_size;
  const float* data1 = (const float*)d_in[0];
  const float* data2 = (const float*)d_in[1];
  const float* dwt   = (const float*)d_in[2];
  const float* d3    = (const float*)d_in[3];
  const float* lin0w = (const float*)d_in[4];
  const float* scw   = (const float*)d_in[5];
  const float* lw0   = (const float*)d_in[6];
  const float* lw1   = (const float*)d_in[7];
  const float* lw2   = (const float*)d_in[8];
  float* out = (float*)d_out;
  float* ws = (float*)d_ws;

  const int N = in_sizes[0] / 576;  // 32768

  hipLaunchKernelGGL(setup_w3j_kernel, dim3(1), dim3(1), 0, stream,
                     (int*)ws, (int2*)(ws + WS_ENT));
  hipLaunchKernelGGL(pack_weights_kernel, dim3(48), dim3(256), 0, stream, lin0w, ws + WS_PL0, 192);
  hipLaunchKernelGGL(pack_weights_kernel, dim3(48), dim3(256), 0, stream, scw,   ws + WS_PSC, 192);
  hipLaunchKernelGGL(pack_weights_kernel, dim3(48), dim3(256), 0, stream, lw0,   ws + WS_PW0, 192);
  hipLaunchKernelGGL(pack_weights_kernel, dim3(64), dim3(256), 0, stream, lw1,   ws + WS_PW1, 256);
  hipLaunchKernelGGL(pack_weights_kernel, dim3(64), dim3(256), 0, stream, lw2,   ws + WS_PW2, 256);
  hipLaunchKernelGGL(fused_tp_kernel, dim3(N / 64), dim3(128), 0, stream,
                     data1, data2, dwt, d3, ws, out);
}